// SomaLayerJAX_2645699854840
// MI455X (gfx1250) — compile-verified
//
#include <hip/hip_runtime.h>

// WICC dynamic-connectivity step for MI455X (gfx1250).
// Memory-bound streaming: 64MB in + 64MB out of edge_state @ 23.3TB/s => ~5.5us floor.
// CDNA5 path: ASYNCcnt-tracked global->LDS b128 copies, double-buffered per wave.

typedef float v4f __attribute__((ext_vector_type(4)));
typedef int   v4i __attribute__((ext_vector_type(4)));
typedef __attribute__((address_space(1))) v4i* gv4p;   // global (AS1) int4*
typedef __attribute__((address_space(3))) v4i* lv4p;   // LDS (AS3) int4*

#if defined(__AMDGCN__) && __has_builtin(__builtin_amdgcn_global_load_async_to_lds_b128)
#define WICC_ASYNC 1
#endif

namespace {
constexpr int   kB            = 16;
constexpr int   kD            = 1024;
constexpr int   kRows         = kB * kD;                // 16384 (b,dst) rows
constexpr int   kRowsPerBlock = 8;                      // same b within a block
constexpr int   kBlocks       = kRows / kRowsPerBlock;  // 2048
constexpr int   kThreads      = 256;                    // 8 wave32 per block
constexpr float kJIn          = 0.38f;
constexpr float kJOut         = 0.38f;
constexpr float kPhiTh        = 0.1675f;
constexpr float kGamma        = 1.0e-3f;
constexpr float kBias         = 2.0f;
}

// 16B global -> LDS copy through the CDNA5 async DMA path (ASYNCcnt-tracked).
__device__ __forceinline__ void async_copy16(const float* gsrc, float* lds_dst) {
#ifdef WICC_ASYNC
  __builtin_amdgcn_global_load_async_to_lds_b128((gv4p)gsrc, (lv4p)lds_dst,
                                                 /*offset=*/0, /*cpol=*/0);
#else
  *reinterpret_cast<v4f*>(lds_dst) = *reinterpret_cast<const v4f*>(gsrc);
#endif
}

template <int N>
__device__ __forceinline__ void wait_async() {
#ifdef WICC_ASYNC
#if __has_builtin(__builtin_amdgcn_s_wait_asynccnt)
  __builtin_amdgcn_s_wait_asynccnt(N);
#else
  asm volatile("s_wait_asynccnt %0" :: "n"(N) : "memory");
#endif
  asm volatile("" ::: "memory");   // keep compiler from hoisting LDS reads above the wait
#endif
}

__device__ __forceinline__ float edge_next(float x, float jv, float s) {
  const float pa = x + jv;
  const float pb = x - jv;
  const float ia = kBias - s;
  const float ib = kBias + s;
  const float ra = __builtin_sqrtf(__builtin_fmaxf(ia * ia - 1.0f, 0.0f));
  const float rb = __builtin_sqrtf(__builtin_fmaxf(ib * ib - 1.0f, 0.0f));
  const float ga = (__builtin_fabsf(pa) > kPhiTh) ? ra : 0.0f;
  const float gb = (__builtin_fabsf(pb) > kPhiTh) ? rb : 0.0f;
  return s + (kGamma * (ga - gb) - kGamma * s);   // DT = 1
}

__global__ __launch_bounds__(kThreads) void wicc_step(
    const float* __restrict__ state,    // [B, D]
    const float* __restrict__ phi,      // [B, D]
    const float* __restrict__ Jm,       // [D, D] (dst, src)
    const float* __restrict__ es,       // [B, D*D]
    float* __restrict__ phi_out,        // [B, D]
    float* __restrict__ es_out) {       // [B, D*D]
  const int t    = threadIdx.x;
  const int row0 = blockIdx.x * kRowsPerBlock;   // row = b*1024 + dst
  const int b    = row0 >> 10;                   // constant within block

  __shared__ __align__(16) float ebuf[2][4 * kThreads];  // double-buffered edge row (2x4KB)
  __shared__ float wsum[kThreads / 32];

  // Source-state row (fixed b), pre-scaled by J_IN; lives in regs for all 8 rows.
  v4f x4 = *reinterpret_cast<const v4f*>(state + ((size_t)b << 10) + 4 * (size_t)t);
  x4 *= kJIn;

  // Prologue: stage row 0's edge slice into LDS buffer 0.   (ASYNCcnt = 1)
  async_copy16(es + ((size_t)row0 << 10) + 4 * (size_t)t, &ebuf[0][4 * t]);

  for (int i = 0; i < kRowsPerBlock; ++i) {
    const int    row   = row0 + i;
    const int    dst   = row & (kD - 1);
    const size_t ebase = (size_t)row << 10;      // contiguous 1024-float edge row

    // Keep next row's DMA in flight while this row computes.
    if (i + 1 < kRowsPerBlock) {
      async_copy16(es + ebase + kD + 4 * (size_t)t, &ebuf[(i + 1) & 1][4 * t]);
      __builtin_prefetch(Jm + (((size_t)dst + 1) << 10) + 4 * (size_t)t, 0, 3);
    }

    // J row: direct load, L2-resident (4MB, reused 16x across batches).
    const v4f j4 = *reinterpret_cast<const v4f*>(Jm + ((size_t)dst << 10) + 4 * (size_t)t);

    // Async loads complete in order per wave: <=1 outstanding leaves only the
    // next row's copy in flight and guarantees this row's LDS data landed.
    if (i + 1 < kRowsPerBlock) wait_async<1>(); else wait_async<0>();

    const v4f s4 = *reinterpret_cast<const v4f*>(&ebuf[i & 1][4 * t]);

    v4f o4;
    o4.x = edge_next(x4.x, j4.x, s4.x);
    o4.y = edge_next(x4.y, j4.y, s4.y);
    o4.z = edge_next(x4.z, j4.z, s4.z);
    o4.w = edge_next(x4.w, j4.w, s4.w);

    // 64MB output stream: non-temporal 128-bit store.
    __builtin_nontemporal_store(o4,
        reinterpret_cast<v4f*>(es_out + ebase + 4 * (size_t)t));

    // Row reduction: wave32 shuffle tree, then combine 8 wave partials via LDS.
    float partial = (o4.x + o4.y) + (o4.z + o4.w);
#pragma unroll
    for (int off = 16; off > 0; off >>= 1)
      partial += __shfl_down(partial, off, 32);
    if ((t & 31) == 0) wsum[t >> 5] = partial;
    __syncthreads();
    if (t == 0) {
      float agg = 0.0f;
#pragma unroll
      for (int w = 0; w < kThreads / 32; ++w) agg += wsum[w];
      phi_out[row] = phi[row] + agg * kJOut;
    }
    __syncthreads();   // protect wsum (and keeps waves in lockstep on ebuf reuse)
  }
}

extern "C" void kernel_launch(void* const* d_in, const int* in_sizes, int n_in,
                              void* d_out, int out_size, void* d_ws, size_t ws_size,
                              hipStream_t stream) {
  (void)in_sizes; (void)n_in; (void)out_size; (void)d_ws; (void)ws_size;
  const float* state = (const float*)d_in[0];   // [16,1024]
  const float* phi   = (const float*)d_in[1];   // [16,1024]
  const float* Jm    = (const float*)d_in[2];   // [1024,1024]
  const float* es    = (const float*)d_in[3];   // [16,1048576]

  float* phi_out = (float*)d_out;                    // output 0: [16,1024]
  float* es_out  = (float*)d_out + (size_t)kB * kD;  // output 1: [16,1048576]

  wicc_step<<<kBlocks, kThreads, 0, stream>>>(state, phi, Jm, es, phi_out, es_out);
}